// RelativeCoordPredictor_74337293959772
// MI455X (gfx1250) — compile-verified
//
#include <hip/hip_runtime.h>
#include <math.h>

typedef float v2f __attribute__((ext_vector_type(2)));
typedef float v8f __attribute__((ext_vector_type(8)));

namespace {
constexpr int C_DIM = 2048;
constexpr int HW = 49;
constexpr int KSTEPS = C_DIM / 4;                         // 512 WMMA K-steps (K=4 each)
constexpr int WAVES_PER_BLOCK = 16;                       // 512 threads, wave32
constexpr int STEPS_PER_WAVE = KSTEPS / WAVES_PER_BLOCK;  // 32
}

// Pack the 16x2048 A matrix (row0=ones, row1=W0hi, row2=W1hi, row3=W0lo, row4=W1lo,
// rows 5..15 = 0) into WMMA f32 16x16x4 A-operand lane layout:
//   lane L: M = L%16 ; VGPR0 holds K = 2*(L/16), VGPR1 holds K = 2*(L/16)+1
// Apack[(step*32 + lane)*2 + j]
__global__ void pack_A(const float* __restrict__ W, float* __restrict__ Apack) {
  int idx = blockIdx.x * blockDim.x + threadIdx.x;
  if (idx >= KSTEPS * 32) return;
  int lane = idx & 31;
  int step = idx >> 5;
  int m = lane & 15;
  int kbase = step * 4 + ((lane >> 4) << 1);
  float v0 = 0.0f, v1 = 0.0f;
#pragma unroll
  for (int j = 0; j < 2; ++j) {
    int kk = kbase + j;
    float val = 0.0f;
    if (m == 0)      val = 1.0f;                 // channel-sum row
    else if (m == 1) val = W[2048 + kk];         // W[0, C+kk]  -> s0
    else if (m == 2) val = W[4096 + 2048 + kk];  // W[1, C+kk]  -> s1
    else if (m == 3) val = W[kk];                // W[0, kk]    -> u0
    else if (m == 4) val = W[4096 + kk];         // W[1, kk]    -> u1
    if (j == 0) v0 = val; else v1 = val;
  }
  float2 packed; packed.x = v0; packed.y = v1;
  ((float2*)Apack)[idx] = packed;
}

__global__ __launch_bounds__(512) void fused(
    const float* __restrict__ x, const float* __restrict__ mask,
    const float* __restrict__ bvec, const float* __restrict__ Apack,
    float* __restrict__ out) {
  __shared__ float part[WAVES_PER_BLOCK][5][64];  // per-wave partial tiles (rows 0..4)
  __shared__ float fin[5][HW];                    // reduced: S, s0, s1, u0, u1
  __shared__ float sm[HW];
  __shared__ float sgap[HW];
  __shared__ int   s_pmax;
  __shared__ float s_gapmean;

  const int tid = threadIdx.x;
  const int n = blockIdx.x;
  const int wave = tid >> 5;
  const int lane = tid & 31;
  const int mrow = lane & 15;
  const int chalf = (lane >> 4) << 1;  // K sub-offset 0 or 2 (A and B share this layout)

  if (tid < HW) sm[tid] = mask[n * HW + tid];

  // B-operand column (position p) per lane for the 4 p-tiles; tile3 clamped to p=48
  const int p0 = mrow, p1 = 16 + mrow, p2 = 32 + mrow;

  v8f a0 = {}, a1 = {}, a2 = {}, a3 = {};
  const float* xn = x + (size_t)n * (C_DIM * HW);
  const int step0 = wave * STEPS_PER_WAVE;

  for (int s = 0; s < STEPS_PER_WAVE; ++s) {
    const int step = step0 + s;
    const float* xr = xn + (size_t)(step * 4 + chalf) * HW;
    v2f A = *(const v2f*)(Apack + 2 * ((size_t)step * 32 + lane));
    v2f B0, B1, B2, B3;
    B0.x = xr[p0];  B0.y = xr[HW + p0];
    B1.x = xr[p1];  B1.y = xr[HW + p1];
    B2.x = xr[p2];  B2.y = xr[HW + p2];
    B3.x = xr[48];  B3.y = xr[HW + 48];
    a0 = __builtin_amdgcn_wmma_f32_16x16x4_f32(false, A, false, B0, (short)0, a0, false, false);
    a1 = __builtin_amdgcn_wmma_f32_16x16x4_f32(false, A, false, B1, (short)0, a1, false, false);
    a2 = __builtin_amdgcn_wmma_f32_16x16x4_f32(false, A, false, B2, (short)0, a2, false, false);
    a3 = __builtin_amdgcn_wmma_f32_16x16x4_f32(false, A, false, B3, (short)0, a3, false, false);
  }

  // D layout: lane<16 + vgpr v -> (M=v, N=lane). We only need rows M=0..4.
  if (lane < 16) {
#pragma unroll
    for (int m = 0; m < 5; ++m) {
      part[wave][m][mrow]      = a0[m];
      part[wave][m][16 + mrow] = a1[m];
      part[wave][m][32 + mrow] = a2[m];
      part[wave][m][48 + mrow] = a3[m];  // only mrow==0 (p=48) is meaningful
    }
  }
  __syncthreads();

  // Deterministic cross-wave reduction (fixed order, no float atomics)
  if (tid < 5 * HW) {
    int m = tid / HW, p = tid % HW;
    float ssum = 0.0f;
    for (int w = 0; w < WAVES_PER_BLOCK; ++w) ssum += part[w][m][p];
    fin[m][p] = ssum;
  }
  __syncthreads();

  // Threshold + argmax (first max, like jnp.argmax)
  if (tid == 0) {
    float tsum = 0.0f;
    for (int p = 0; p < HW; ++p) tsum += sm[p];
    const float thr = tsum / 49.0f;
    int pmax = 0;
    float best = -__builtin_inff();
    for (int p = 0; p < HW; ++p) {
      float mm = (sm[p] > thr) ? (fin[0][p] / 2048.0f) : 0.0f;
      if (mm > best) { best = mm; pmax = p; }
    }
    s_pmax = pmax;
  }
  __syncthreads();

  const int pmax = s_pmax;
  float pred_d = 0.0f, rdist = 0.0f;
  if (tid < HW) {
    const float u0 = fin[3][pmax];
    const float u1 = fin[4][pmax];
    float pd = fmaxf(u0 + fin[1][tid] + bvec[0], 0.0f);
    float pa = fmaxf(u1 + fin[2][tid] + bvec[1], 0.0f);
    if (tid == pmax) { pd = 0.0f; pa = 0.0f; }
    const int ii = tid / 7,  jj = tid % 7;
    const int i0 = pmax / 7, j0 = pmax % 7;
    const float rx = (float)(ii - i0) / 7.0f;
    const float ry = (float)(jj - j0) / 7.0f;
    rdist = sqrtf(rx * rx + ry * ry);
    const float ang = (atan2f(ry, rx) / 3.14159265358979323846f + 1.0f) * 0.5f;
    float gap = pa - ang;
    if (gap < 0.0f) gap += 1.0f;
    sgap[tid] = gap;
    pred_d = pd;
  }
  __syncthreads();
  if (tid == 0) {
    float gs = 0.0f;
    for (int p = 0; p < HW; ++p) gs += sgap[p];
    s_gapmean = gs / 49.0f;
  }
  __syncthreads();
  if (tid < HW) {
    const float g = sgap[tid] - s_gapmean;
    const float dl = pred_d - rdist;
    out[(size_t)n * HW + tid] = dl * dl + g * g;
  }
}

extern "C" void kernel_launch(void* const* d_in, const int* in_sizes, int n_in,
                              void* d_out, int out_size, void* d_ws, size_t ws_size,
                              hipStream_t stream) {
  const float* x    = (const float*)d_in[0];  // (256, 2048, 7, 7)
  const float* mask = (const float*)d_in[1];  // (256, 7, 7)
  const float* W    = (const float*)d_in[2];  // (2, 4096)
  const float* b    = (const float*)d_in[3];  // (2,)
  float* out = (float*)d_out;                 // (256, 7, 7)
  float* Apack = (float*)d_ws;                // 512*32*2 floats = 128 KB

  pack_A<<<(KSTEPS * 32 + 255) / 256, 256, 0, stream>>>(W, Apack);
  fused<<<256, 512, 0, stream>>>(x, mask, b, Apack, out);
}